// GIN_62646392980003
// MI455X (gfx1250) — compile-verified
//
#include <hip/hip_runtime.h>
#include <hip/hip_bf16.h>

typedef __attribute__((ext_vector_type(2))) float v2f;
typedef __attribute__((ext_vector_type(8))) float v8f;

#define DIM 64
#define WPAD 68   // padded LDS row stride (floats): bank = (4*row + k) % 64 -> conflict-free b64

// ---------------------------------------------------------------- utilities
__global__ void gin_copy_f4(const float* __restrict__ in, float* __restrict__ outp, int n4) {
    int i = blockIdx.x * blockDim.x + threadIdx.x;
    if (i < n4) ((float4*)outp)[i] = ((const float4*)in)[i];
}

__global__ void gin_zero_f(float* __restrict__ p, int n) {
    int i = blockIdx.x * blockDim.x + threadIdx.x;
    if (i < n) p[i] = 0.0f;
}

// ------------------------------------------------- edge scatter-add (GIN agg)
// 16 threads per edge; each handles 4 of the 64 features (float4 gather, 4 atomics).
__global__ void gin_scatter(const float* __restrict__ h, const int* __restrict__ src,
                            const int* __restrict__ dst, float* __restrict__ z, int nEdges) {
    long long t = (long long)blockIdx.x * blockDim.x + threadIdx.x;
    int e = (int)(t >> 4);
    int q = (int)(t & 15);
    if (e >= nEdges) return;
    int s = src[e];
    int d = dst[e];
    float4 v = *(const float4*)(h + (size_t)s * DIM + q * 4);
    float* zp = z + (size_t)d * DIM + q * 4;
    atomicAdd(zp + 0, v.x);
    atomicAdd(zp + 1, v.y);
    atomicAdd(zp + 2, v.z);
    atomicAdd(zp + 3, v.w);
}

// ------------------------------------------------- graph mean-pool pieces
__global__ void gin_pool_scatter(const float* __restrict__ h, const int* __restrict__ batch,
                                 float* __restrict__ sums, float* __restrict__ cnts, int nNodes) {
    long long t = (long long)blockIdx.x * blockDim.x + threadIdx.x;
    int node = (int)(t >> 4);
    int q = (int)(t & 15);
    if (node >= nNodes) return;
    int g = batch[node];
    float4 v = *(const float4*)(h + (size_t)node * DIM + q * 4);
    float* sp = sums + (size_t)g * DIM + q * 4;
    atomicAdd(sp + 0, v.x);
    atomicAdd(sp + 1, v.y);
    atomicAdd(sp + 2, v.z);
    atomicAdd(sp + 3, v.w);
    if (q == 0) atomicAdd(&cnts[g], 1.0f);
}

__global__ void gin_pool_finalize(const float* __restrict__ sums, const float* __restrict__ cnts,
                                  float* __restrict__ pooled, int n) {
    int i = blockIdx.x * blockDim.x + threadIdx.x;
    if (i < n) {
        float c = cnts[i >> 6];
        pooled[i] = sums[i] / fmaxf(c, 1.0f);
    }
}

// ------------------------------------------------- fused MLP via f32 WMMA
// Xout = relu(Xin @ W1 + b1) @ W2 + b2, M rows x 64 features.
// 8 wave32 per workgroup, each wave owns a 16-row tile.
// V_WMMA_F32_16X16X4_F32 layouts (ISA 7.12.2):
//   A (16x4 f32): lanes 0-15 -> M=lane, VGPR{0,1}=K{0,1}; lanes 16-31 -> K{2,3}.
//   B (4x16 f32): lane half selects K pair, VGPR index selects K within pair, N=lane%16.
//   C/D (16x16 f32): VGPR r: lanes 0-15 -> M=r, lanes 16-31 -> M=r+8; N=lane%16.
// Weights are staged TRANSPOSED in LDS (sWT[n][k], stride WPAD) so each B fragment
// is one contiguous 8B ds_load_b64 into an even-aligned VGPR pair, bank-conflict-free.
__launch_bounds__(256)
__global__ void gin_mlp_wmma(const float* __restrict__ Xin,
                             const float* __restrict__ W1, const float* __restrict__ b1,
                             const float* __restrict__ W2, const float* __restrict__ b2,
                             float* __restrict__ Xout, int M) {
    __shared__ float sW1T[DIM * WPAD];
    __shared__ float sW2T[DIM * WPAD];
    __shared__ float sB[2 * DIM];
    __shared__ float sZ[8][16 * WPAD];   // per-wave relu(Z@W1+b1) staging, padded

    const int tid = threadIdx.x;
    // cooperative transpose-load of the two 64x64 weight matrices
    for (int i = tid; i < (DIM * DIM) / 4; i += 256) {
        int k = i >> 4;              // source row (K index)
        int n0 = (i & 15) * 4;       // source col group (N index)
        float4 w1 = ((const float4*)W1)[i];
        float4 w2 = ((const float4*)W2)[i];
        sW1T[(n0 + 0) * WPAD + k] = w1.x;
        sW1T[(n0 + 1) * WPAD + k] = w1.y;
        sW1T[(n0 + 2) * WPAD + k] = w1.z;
        sW1T[(n0 + 3) * WPAD + k] = w1.w;
        sW2T[(n0 + 0) * WPAD + k] = w2.x;
        sW2T[(n0 + 1) * WPAD + k] = w2.y;
        sW2T[(n0 + 2) * WPAD + k] = w2.z;
        sW2T[(n0 + 3) * WPAD + k] = w2.w;
    }
    if (tid < DIM) { sB[tid] = b1[tid]; sB[DIM + tid] = b2[tid]; }
    __syncthreads();

    const int wave = tid >> 5;
    const int lane = tid & 31;
    const int half = lane >> 4;   // 0: K{0,1}, 1: K{2,3} within each K-block of 4
    const int ln   = lane & 15;
    const int rowBase = blockIdx.x * 128 + wave * 16;

    // ---- GEMM1: acc = Z @ W1 + b1 ------------------------------------
    v8f acc[4];
    for (int nt = 0; nt < 4; ++nt) {
        float bv = sB[nt * 16 + ln];
        for (int r = 0; r < 8; ++r) acc[nt][r] = bv;
    }
    int arow = rowBase + ln;
    if (arow >= M) arow = M - 1;              // clamp loads; EXEC stays all-ones for WMMA
    const float* aptr = Xin + (size_t)arow * DIM + half * 2;

    for (int kb = 0; kb < 16; ++kb) {
        v2f a = *(const v2f*)(aptr + kb * 4);         // K = 4kb + 2*half {+0,+1}
        const int k0 = kb * 4 + half * 2;
        for (int nt = 0; nt < 4; ++nt) {
            v2f b = *(const v2f*)(&sW1T[(nt * 16 + ln) * WPAD + k0]);  // ds_load_b64
            acc[nt] = __builtin_amdgcn_wmma_f32_16x16x4_f32(
                false, a, false, b, (short)0, acc[nt], false, false);
        }
    }

    // ---- ReLU + restage through per-wave LDS (re-form A fragments) ----
    float* zb = sZ[wave];
    for (int nt = 0; nt < 4; ++nt)
        for (int r = 0; r < 8; ++r) {
            float v = acc[nt][r];
            zb[(r + half * 8) * WPAD + nt * 16 + ln] = v > 0.0f ? v : 0.0f;
        }
    // LDS ops from one wave complete in order (DScnt, in-order per ISA); wave-private buffer.

    // ---- GEMM2: out = relu() @ W2 + b2 -------------------------------
    v8f acc2[4];
    for (int nt = 0; nt < 4; ++nt) {
        float bv = sB[DIM + nt * 16 + ln];
        for (int r = 0; r < 8; ++r) acc2[nt][r] = bv;
    }
    for (int kb = 0; kb < 16; ++kb) {
        const int k0 = kb * 4 + half * 2;
        v2f a = *(const v2f*)(&zb[ln * WPAD + k0]);                    // ds_load_b64
        for (int nt = 0; nt < 4; ++nt) {
            v2f b = *(const v2f*)(&sW2T[(nt * 16 + ln) * WPAD + k0]);  // ds_load_b64
            acc2[nt] = __builtin_amdgcn_wmma_f32_16x16x4_f32(
                false, a, false, b, (short)0, acc2[nt], false, false);
        }
    }

    // ---- guarded store ------------------------------------------------
    for (int nt = 0; nt < 4; ++nt)
        for (int r = 0; r < 8; ++r) {
            int row = rowBase + r + half * 8;
            if (row < M) Xout[(size_t)row * DIM + nt * 16 + ln] = acc2[nt][r];
        }
}

// ---------------------------------------------------------------- launcher
extern "C" void kernel_launch(void* const* d_in, const int* in_sizes, int n_in,
                              void* d_out, int out_size, void* d_ws, size_t ws_size,
                              hipStream_t stream) {
    const float* x     = (const float*)d_in[0];
    const int*   eidx  = (const int*)d_in[1];
    const int*   batch = (const int*)d_in[2];
    const float* Ws1   = (const float*)d_in[3];
    const float* bs1   = (const float*)d_in[4];
    const float* Ws2   = (const float*)d_in[5];
    const float* bs2   = (const float*)d_in[6];
    const float* mW1   = (const float*)d_in[7];
    const float* mb1   = (const float*)d_in[8];
    const float* mW2   = (const float*)d_in[9];
    const float* mb2   = (const float*)d_in[10];
    float* out = (float*)d_out;

    const int nNodes  = in_sizes[0] / DIM;   // 100000
    const int nEdges  = in_sizes[1] / 2;     // 1600000
    const int nGraphs = out_size / DIM;      // 256
    const int nLayers = in_sizes[3] / (DIM * DIM);  // 5

    const int* src = eidx;
    const int* dst = eidx + nEdges;

    float* B0     = (float*)d_ws;                       // h ping
    float* B1     = B0 + (size_t)nNodes * DIM;          // z (agg target)
    float* sums   = B1 + (size_t)nNodes * DIM;          // [nGraphs, 64]
    float* cnts   = sums + (size_t)nGraphs * DIM;       // [nGraphs]
    float* pooled = cnts + nGraphs;                     // [nGraphs, 64]

    const dim3 blk(256);
    const float* hcur = x;
    const int n4 = nNodes * (DIM / 4);
    const long long escatter = (long long)nEdges * 16;

    for (int l = 0; l < nLayers; ++l) {
        // z = h  (self term of GIN, eps=0)
        gin_copy_f4<<<(n4 + 255) / 256, blk, 0, stream>>>(hcur, B1, n4);
        // z[dst] += h[src]
        gin_scatter<<<(unsigned)((escatter + 255) / 256), blk, 0, stream>>>(hcur, src, dst, B1, nEdges);
        // h = relu(z@W1+b1)@W2+b2
        gin_mlp_wmma<<<(nNodes + 127) / 128, blk, 0, stream>>>(
            B1, Ws1 + (size_t)l * DIM * DIM, bs1 + (size_t)l * DIM,
            Ws2 + (size_t)l * DIM * DIM, bs2 + (size_t)l * DIM, B0, nNodes);
        hcur = B0;
    }

    // global mean pool
    const int zn = nGraphs * DIM + nGraphs;  // sums and cnts are contiguous
    gin_zero_f<<<(zn + 255) / 256, blk, 0, stream>>>(sums, zn);
    gin_pool_scatter<<<(unsigned)(((long long)nNodes * 16 + 255) / 256), blk, 0, stream>>>(
        hcur, batch, sums, cnts, nNodes);
    gin_pool_finalize<<<(nGraphs * DIM + 255) / 256, blk, 0, stream>>>(
        sums, cnts, pooled, nGraphs * DIM);

    // readout MLP into d_out
    gin_mlp_wmma<<<(nGraphs + 127) / 128, blk, 0, stream>>>(
        pooled, mW1, mb1, mW2, mb2, out, nGraphs);
}